// RollingShutterDetector_38809324487127
// MI455X (gfx1250) — compile-verified
//
#include <hip/hip_runtime.h>
#include <math.h>

typedef float v2f __attribute__((ext_vector_type(2)));
typedef float v8f __attribute__((ext_vector_type(8)));

#define BB   8
#define TT   16
#define HH   512
#define WW   512
#define PP   15
#define RB   64
#define NBLK (HH / RB)

// ---------------------------------------------------------------------------
// Kernel 1: fused streaming stats. One WG = (b, pair, 64-row block).
// 128 threads, 4 columns/thread via float4 (b128) loads.
// Writes per-block partials: hd_part/edge_part [bp][blk][512], vd_part[bp][blk].
// ---------------------------------------------------------------------------
__global__ __launch_bounds__(128) void rs_pair_stats(
    const float* __restrict__ frames,
    float* __restrict__ hd_part,
    float* __restrict__ edge_part,
    float* __restrict__ vd_part)
{
    __shared__ alignas(16) float s_img[3][WW];   // uint8-quantized frame t+1, 3-row ring
    __shared__ float s_red[128];

    const int tid = threadIdx.x;
    const int wg  = blockIdx.x;                  // ((b*PP)+p)*NBLK + blk
    const int blk = wg % NBLK;
    const int bp  = wg / NBLK;
    const int p   = bp % PP;
    const int b   = bp / PP;
    const int c0  = tid * 4;
    const int r0  = blk * RB;
    const int r1  = r0 + RB;

    const size_t fsz = (size_t)HH * WW;
    const float* fA = frames + (size_t)((b * TT + p) * 3) * fsz;  // frame t
    const float* fB = fA + 3 * fsz;                                // frame t+1

    float4 hd_acc = make_float4(0.f, 0.f, 0.f, 0.f);
    float4 ec     = make_float4(0.f, 0.f, 0.f, 0.f);
    float  vd_acc = 0.f;

    // Load one row: gray both frames, d = grayB - grayA (returned),
    // img = floor(clip(grayB*255,0,255)) -> LDS ring slot s.
    auto loadRow = [&](int h, int s) -> float4 {
        const size_t off = (size_t)h * WW + c0;
        float4 a0 = *(const float4*)(fA + off);
        float4 a1 = *(const float4*)(fA + fsz + off);
        float4 a2 = *(const float4*)(fA + 2 * fsz + off);
        float4 q0 = *(const float4*)(fB + off);
        float4 q1 = *(const float4*)(fB + fsz + off);
        float4 q2 = *(const float4*)(fB + 2 * fsz + off);
        if (h + 8 < HH) {                        // software prefetch 8 rows ahead
            const size_t poff = off + (size_t)8 * WW;
            __builtin_prefetch(fA + poff, 0, 3);
            __builtin_prefetch(fA + fsz + poff, 0, 3);
            __builtin_prefetch(fA + 2 * fsz + poff, 0, 3);
            __builtin_prefetch(fB + poff, 0, 3);
            __builtin_prefetch(fB + fsz + poff, 0, 3);
            __builtin_prefetch(fB + 2 * fsz + poff, 0, 3);
        }
        const float k3 = (1.0f / 3.0f);
        float4 d, im;
        float gB;
        gB   = (q0.x + q1.x + q2.x) * k3;
        d.x  = gB - (a0.x + a1.x + a2.x) * k3;
        im.x = floorf(fminf(fmaxf(gB * 255.f, 0.f), 255.f));
        gB   = (q0.y + q1.y + q2.y) * k3;
        d.y  = gB - (a0.y + a1.y + a2.y) * k3;
        im.y = floorf(fminf(fmaxf(gB * 255.f, 0.f), 255.f));
        gB   = (q0.z + q1.z + q2.z) * k3;
        d.z  = gB - (a0.z + a1.z + a2.z) * k3;
        im.z = floorf(fminf(fmaxf(gB * 255.f, 0.f), 255.f));
        gB   = (q0.w + q1.w + q2.w) * k3;
        d.w  = gB - (a0.w + a1.w + a2.w) * k3;
        im.w = floorf(fminf(fmaxf(gB * 255.f, 0.f), 255.f));
        *(float4*)(&s_img[s][c0]) = im;
        return d;
    };

    int sp = 0, sc = 1, sn = 2;                  // ring slots prev/cur/next
    float4 d_prev = make_float4(0.f, 0.f, 0.f, 0.f);
    float4 d_cur, d_next = make_float4(0.f, 0.f, 0.f, 0.f);
    if (r0 > 0) d_prev = loadRow(r0 - 1, sp);    // halo row
    d_cur = loadRow(r0, sc);

    for (int h = r0; h < r1; ++h) {
        if (h + 1 < HH) d_next = loadRow(h + 1, sn);
        __syncthreads();                          // LDS row(s) visible

        // vd: row-gradient of d (central diff interior, one-sided at edges)
        float4 gd;
        if (h == 0) {
            gd.x = d_next.x - d_cur.x; gd.y = d_next.y - d_cur.y;
            gd.z = d_next.z - d_cur.z; gd.w = d_next.w - d_cur.w;
        } else if (h == HH - 1) {
            gd.x = d_cur.x - d_prev.x; gd.y = d_cur.y - d_prev.y;
            gd.z = d_cur.z - d_prev.z; gd.w = d_cur.w - d_prev.w;
        } else {
            gd.x = (d_next.x - d_prev.x) * 0.5f; gd.y = (d_next.y - d_prev.y) * 0.5f;
            gd.z = (d_next.z - d_prev.z) * 0.5f; gd.w = (d_next.w - d_prev.w) * 0.5f;
        }
        vd_acc += fabsf(gd.x) + fabsf(gd.y) + fabsf(gd.z) + fabsf(gd.w);

        // hd: per-column sum of |d|
        hd_acc.x += fabsf(d_cur.x); hd_acc.y += fabsf(d_cur.y);
        hd_acc.z += fabsf(d_cur.z); hd_acc.w += fabsf(d_cur.w);

        // Sobel on quantized frame t+1 with reflect-101 borders
        const int rp = (h == 0) ? sn : sp;
        const int rn = (h == HH - 1) ? sp : sn;
        const int rc = sc;
        float* eca = &ec.x;
        #pragma unroll
        for (int j = 0; j < 4; ++j) {
            const int c  = c0 + j;
            const int cm = (c == 0) ? 1 : (c - 1);
            const int cp = (c == WW - 1) ? (WW - 2) : (c + 1);
            const float a00 = s_img[rp][cm], a01 = s_img[rp][c], a02 = s_img[rp][cp];
            const float a10 = s_img[rc][cm],                     a12 = s_img[rc][cp];
            const float a20 = s_img[rn][cm], a21 = s_img[rn][c], a22 = s_img[rn][cp];
            const float gx = (a02 - a00) + 2.f * (a12 - a10) + (a22 - a20);
            const float gy = (a20 - a00) + 2.f * (a21 - a01) + (a22 - a02);
            const float mag = fabsf(gx) + fabsf(gy);
            eca[j] += (mag > 150.f) ? 255.f : 0.f;
        }
        __syncthreads();                          // reads done before slot reuse
        const int t = sp; sp = sc; sc = sn; sn = t;
        d_prev = d_cur; d_cur = d_next;
    }

    // per-column partials (deterministic: own slot, no atomics)
    *(float4*)(hd_part   + (size_t)wg * WW + c0) = hd_acc;
    *(float4*)(edge_part + (size_t)wg * WW + c0) = ec;

    // block-wide vd partial via fixed-order tree reduction
    s_red[tid] = vd_acc;
    __syncthreads();
    for (int s = 64; s > 0; s >>= 1) {
        if (tid < s) s_red[tid] += s_red[tid + s];
        __syncthreads();
    }
    if (tid == 0) vd_part[wg] = s_red[0];
}

// ---------------------------------------------------------------------------
// Kernel 2: per-(b,pair) finalize -> [vd, hm, ratio, wobble]
// ---------------------------------------------------------------------------
__global__ __launch_bounds__(256) void rs_pair_finalize(
    const float* __restrict__ hd_part,
    const float* __restrict__ edge_part,
    const float* __restrict__ vd_part,
    float* __restrict__ pair_stats)
{
    __shared__ float sA[256], sB[256], sC[256], sD[256];
    const int bp  = blockIdx.x;
    const int tid = threadIdx.x;

    float sh = 0.f, sh2 = 0.f, scl = 0.f, scl2 = 0.f;
    for (int c = tid; c < WW; c += 256) {
        float hv = 0.f, cv = 0.f;
        #pragma unroll
        for (int k = 0; k < NBLK; ++k) {
            hv += hd_part  [(size_t)(bp * NBLK + k) * WW + c];
            cv += edge_part[(size_t)(bp * NBLK + k) * WW + c];
        }
        hv *= (1.0f / HH);                        // mean over H
        sh += hv; sh2 += hv * hv;
        scl += cv; scl2 += cv * cv;
    }
    sA[tid] = sh; sB[tid] = sh2; sC[tid] = scl; sD[tid] = scl2;
    __syncthreads();
    for (int s = 128; s > 0; s >>= 1) {
        if (tid < s) {
            sA[tid] += sA[tid + s]; sB[tid] += sB[tid + s];
            sC[tid] += sC[tid + s]; sD[tid] += sD[tid + s];
        }
        __syncthreads();
    }
    if (tid == 0) {
        float vd = 0.f;
        #pragma unroll
        for (int k = 0; k < NBLK; ++k) vd += vd_part[bp * NBLK + k];
        vd *= 1.0f / ((float)HH * (float)WW);
        const float mh  = sA[0] * (1.0f / WW);
        const float hm  = sqrtf(fmaxf(sB[0] * (1.0f / WW) - mh * mh, 0.f));
        const float mc  = sC[0] * (1.0f / WW);
        const float wob = sqrtf(fmaxf(sD[0] * (1.0f / WW) - mc * mc, 0.f));
        const float ratio = (hm > 0.f) ? (vd / (hm + 1e-6f)) : 0.f;
        pair_stats[bp * 4 + 0] = vd;
        pair_stats[bp * 4 + 1] = hm;
        pair_stats[bp * 4 + 2] = ratio;
        pair_stats[bp * 4 + 3] = wob;
    }
}

// ---------------------------------------------------------------------------
// Kernel 3: single wave32 — stats mean over pairs + MLP via V_WMMA_F32_16X16X4_F32
//   GEMM1: [16(pad8),4] @ [4,16(pad8)] + b1 -> relu
//   GEMM2: [16(pad8),8] @ [8,16] + b2, K=8 as two accumulating K=4 WMMAs
// ---------------------------------------------------------------------------
__global__ __launch_bounds__(32) void rs_mlp(
    const float* __restrict__ pair_stats,
    const float* __restrict__ W1, const float* __restrict__ b1,
    const float* __restrict__ W2, const float* __restrict__ b2,
    float* __restrict__ out)
{
    __shared__ float s_stats[8][4];
    __shared__ float s_h[8][16];
    const int L = threadIdx.x;

    // stats[b][j] = mean over 15 pairs (fixed order)
    {
        const int bb = L >> 2, j = L & 3;
        float acc = 0.f;
        #pragma unroll
        for (int p = 0; p < PP; ++p) acc += pair_stats[(bb * PP + p) * 4 + j];
        s_stats[bb][j] = acc * (1.0f / PP);
    }
    __syncthreads();

    // A/B fragment mapping for 16x16x4 f32 (ISA 7.12.2):
    // lane L: M/N = L&15 ; lanes 0-15 hold K=0,1 in v[0],v[1]; lanes 16-31 hold K=2,3
    const int MN = L & 15;
    const int K0 = (L >> 4) * 2;

    v2f a1, bf1;
    a1.x  = (MN < 8) ? s_stats[MN][K0]     : 0.f;
    a1.y  = (MN < 8) ? s_stats[MN][K0 + 1] : 0.f;
    bf1.x = (MN < 8) ? W1[K0 * 8 + MN]       : 0.f;   // W1[k][n], [4,8]
    bf1.y = (MN < 8) ? W1[(K0 + 1) * 8 + MN] : 0.f;
    const float c1v = (MN < 8) ? b1[MN] : 0.f;
    v8f acc1 = {c1v, c1v, c1v, c1v, c1v, c1v, c1v, c1v};

    acc1 = __builtin_amdgcn_wmma_f32_16x16x4_f32(
        false, a1, false, bf1, (short)0, acc1, false, false);

    // relu; keep rows 0..7 (lanes 0-15, reg r = row r, col = lane)
    if (L < 16) {
        #pragma unroll
        for (int r = 0; r < 8; ++r) s_h[r][L] = fmaxf(acc1[r], 0.f);
    }
    __syncthreads();

    v2f a2c0, a2c1, b2c0, b2c1;
    a2c0.x = (MN < 8) ? s_h[MN][K0]     : 0.f;
    a2c0.y = (MN < 8) ? s_h[MN][K0 + 1] : 0.f;
    a2c1.x = (MN < 8) ? s_h[MN][K0 + 4] : 0.f;
    a2c1.y = (MN < 8) ? s_h[MN][K0 + 5] : 0.f;
    b2c0.x = W2[K0 * 16 + MN];                        // W2[k][n], [8,16]
    b2c0.y = W2[(K0 + 1) * 16 + MN];
    b2c1.x = W2[(K0 + 4) * 16 + MN];
    b2c1.y = W2[(K0 + 5) * 16 + MN];
    const float c2v = b2[MN];
    v8f acc2 = {c2v, c2v, c2v, c2v, c2v, c2v, c2v, c2v};

    acc2 = __builtin_amdgcn_wmma_f32_16x16x4_f32(
        false, a2c0, false, b2c0, (short)0, acc2, false, false);
    acc2 = __builtin_amdgcn_wmma_f32_16x16x4_f32(
        false, a2c1, false, b2c1, (short)0, acc2, false, false);

    if (L < 16) {
        #pragma unroll
        for (int r = 0; r < 8; ++r) out[r * 16 + L] = acc2[r];   // out[8][16]
    }
}

// ---------------------------------------------------------------------------
extern "C" void kernel_launch(void* const* d_in, const int* in_sizes, int n_in,
                              void* d_out, int out_size, void* d_ws, size_t ws_size,
                              hipStream_t stream) {
    (void)in_sizes; (void)n_in; (void)out_size; (void)ws_size;
    const float* frames = (const float*)d_in[0];
    const float* W1     = (const float*)d_in[1];
    const float* b1     = (const float*)d_in[2];
    const float* W2     = (const float*)d_in[3];
    const float* b2     = (const float*)d_in[4];
    float* out = (float*)d_out;

    float* ws         = (float*)d_ws;
    float* hd_part    = ws;                                   // 8*15*8*512
    float* edge_part  = hd_part   + (size_t)BB * PP * NBLK * WW;
    float* vd_part    = edge_part + (size_t)BB * PP * NBLK * WW;  // 960
    float* pair_stats = vd_part   + (size_t)BB * PP * NBLK;       // 480

    rs_pair_stats   <<<dim3(BB * PP * NBLK), dim3(128), 0, stream>>>(
        frames, hd_part, edge_part, vd_part);
    rs_pair_finalize<<<dim3(BB * PP),        dim3(256), 0, stream>>>(
        hd_part, edge_part, vd_part, pair_stats);
    rs_mlp          <<<dim3(1),              dim3(32),  0, stream>>>(
        pair_stats, W1, b1, W2, b2, out);
}